// ModelNew_7868380086956
// MI455X (gfx1250) — compile-verified
//
#include <hip/hip_runtime.h>
#include <stdint.h>

// MoE token scatter: out[expert_offsets[expert_idx[t]] + slot_idx[t], :] = token_hidden[t, :]
// Pure streaming permutation: 512 MiB traffic, 0 FLOPs -> ~23 us floor @ 23.3 TB/s.
// Uses the CDNA5 async global<->LDS DMA path (ASYNCcnt) so data never stages in VGPRs.

#define HIDDEN          1024
#define ROW_BYTES       (HIDDEN * 4)      // 4096 B per row
#define ROWS_PER_BLOCK  8                 // 32 KiB LDS bounce buffer
#define THREADS         256               // 8 waves (wave32); 256*16B == one row

#define AS1 __attribute__((address_space(1)))
#define AS3 __attribute__((address_space(3)))

typedef __attribute__((ext_vector_type(4))) int v4i;

__device__ __forceinline__ void async_g2l_b128(const void* gsrc, void* lds) {
#if __has_builtin(__builtin_amdgcn_global_load_async_to_lds_b128)
  __builtin_amdgcn_global_load_async_to_lds_b128((AS1 v4i*)gsrc, (AS3 v4i*)lds,
                                                 /*offset=*/0, /*cpol=*/0);
#else
  uint32_t laddr = (uint32_t)(uintptr_t)lds;          // flat LDS addr low 32 bits = LDS offset
  asm volatile("global_load_async_to_lds_b128 %0, %1, off"
               :: "v"(laddr), "v"(gsrc) : "memory");
#endif
}

__device__ __forceinline__ void async_l2g_b128(void* gdst, const void* lds) {
#if __has_builtin(__builtin_amdgcn_global_store_async_from_lds_b128)
  __builtin_amdgcn_global_store_async_from_lds_b128((AS1 v4i*)gdst, (AS3 v4i*)lds,
                                                    /*offset=*/0, /*cpol=*/0);
#else
  uint32_t laddr = (uint32_t)(uintptr_t)lds;
  asm volatile("global_store_async_from_lds_b128 %0, %1, off"
               :: "v"(gdst), "v"(laddr) : "memory");
#endif
}

__device__ __forceinline__ void wait_async0() {
#if __has_builtin(__builtin_amdgcn_s_wait_asynccnt)
  __builtin_amdgcn_s_wait_asynccnt(0);
#else
  asm volatile("s_wait_asynccnt 0x0" ::: "memory");
#endif
}

__global__ __launch_bounds__(THREADS) void moe_scatter_rows_async(
    const float* __restrict__ src,        // [num_tokens, HIDDEN]
    const int*   __restrict__ expert_idx, // [num_tokens]
    const int*   __restrict__ slot_idx,   // [num_tokens]
    const int*   __restrict__ expert_off, // [E+1]
    float*       __restrict__ dst,        // [rows, HIDDEN], rows == num_tokens
    int num_tokens)
{
  __shared__ alignas(16) char lbuf[ROWS_PER_BLOCK * ROW_BYTES];  // 32 KiB
  const int  tid  = threadIdx.x;
  const long base = (long)blockIdx.x * ROWS_PER_BLOCK;

  // Destination rows for this block's tokens (uniform across the block; tiny
  // broadcast loads that hit L2/L0).
  long destrow[ROWS_PER_BLOCK];
#pragma unroll
  for (int r = 0; r < ROWS_PER_BLOCK; ++r) {
    long t = base + r;
    destrow[r] = (t < (long)num_tokens)
                   ? (long)expert_off[expert_idx[t]] + (long)slot_idx[t]
                   : -1;
  }

  // Phase 1: 8 outstanding async loads per thread, global -> LDS.
  // Each thread owns bytes [tid*16, tid*16+16) of each row.
#pragma unroll
  for (int r = 0; r < ROWS_PER_BLOCK; ++r) {
    long t = base + r;
    if (t < (long)num_tokens) {
      async_g2l_b128(src + t * HIDDEN + (long)tid * 4,
                     lbuf + r * ROW_BYTES + tid * 16);
    }
  }

  // Per-wave ASYNCcnt drain: guarantees this wave's LDS bytes are written.
  // Each lane only stores what it loaded, so no cross-wave barrier is needed.
  wait_async0();

  // Phase 2: 8 outstanding async stores, LDS -> permuted global rows.
  // S_ENDPGM's implicit wait-idle drains the stores.
#pragma unroll
  for (int r = 0; r < ROWS_PER_BLOCK; ++r) {
    if (destrow[r] >= 0) {
      async_l2g_b128(dst + destrow[r] * HIDDEN + (long)tid * 4,
                     lbuf + r * ROW_BYTES + tid * 16);
    }
  }
}

extern "C" void kernel_launch(void* const* d_in, const int* in_sizes, int n_in,
                              void* d_out, int out_size, void* d_ws, size_t ws_size,
                              hipStream_t stream) {
  (void)n_in; (void)out_size; (void)d_ws; (void)ws_size;
  const float* token_hidden = (const float*)d_in[0];
  const int*   expert_idx   = (const int*)d_in[1];
  const int*   slot_idx     = (const int*)d_in[2];
  const int*   expert_off   = (const int*)d_in[3];
  float*       out          = (float*)d_out;

  const int num_tokens = in_sizes[1];  // 65536
  const int nblocks = (num_tokens + ROWS_PER_BLOCK - 1) / ROWS_PER_BLOCK;

  moe_scatter_rows_async<<<nblocks, THREADS, 0, stream>>>(
      token_hidden, expert_idx, slot_idx, expert_off, out, num_tokens);
}